// SineGen_46694884442164
// MI455X (gfx1250) — compile-verified
//
#include <hip/hip_runtime.h>
#include <hip/hip_bf16.h>
#include <math.h>

// ---------------- SineGen config (matches reference) ----------------
#define FS        24000.0f
#define DIM       9
#define BATCH     16
#define TFRAMES   500
#define UPP       256
#define SAMPLES   (TFRAMES * UPP)          // 128000
#define AMP       0.1f
#define NOISE_STD 0.003f

// flat output offsets (sine | uv | noise)
#define UV_OFF    ((size_t)BATCH * SAMPLES * DIM)            // 18,432,000
#define NOISE_OFF (UV_OFF + (size_t)BATCH * SAMPLES)          // 20,480,000

// ---------------------------------------------------------------------------
// Kernel A: per-(b,d) frame-level scan (T=500).
//   rad[b,t,d]  = frac(f0[b,t]*(d+1)/FS)  (+ rand_ini at t==0, harmonic 0 zeroed)
//   base[b,t,d] = 256 * sum_{tau<t} rad[b,tau,d]   (f64 — phase reaches ~6e4)
// Tiny: 144 chains x 500 sequential f64 adds.
// ---------------------------------------------------------------------------
__global__ void sinegen_frames(const float* __restrict__ f0,
                               const float* __restrict__ rand_ini,
                               double* __restrict__ base,
                               float* __restrict__ radf) {
    const int b = blockIdx.x;
    const int d = threadIdx.x;
    if (d >= DIM) return;

    const float harm = (float)(d + 1);
    double csum = 0.0;
    for (int t = 0; t < TFRAMES; ++t) {
        float fn = f0[b * TFRAMES + t] * harm;   // f32, as reference
        float r  = fn / FS;
        r = r - floorf(r);                        // % 1.0 (x >= 0)
        if (t == 0 && d > 0) r += rand_ini[b * DIM + d];
        const int idx = (b * TFRAMES + t) * DIM + d;
        base[idx] = 256.0 * csum;                 // phase base BEFORE frame t
        radf[idx] = r;
        csum += (double)r;                        // exact-enough f64 accumulation
    }
}

// ---------------------------------------------------------------------------
// Kernel B: one block per (b, frame); one thread per sample (k = 0..255).
//   phase(s=t*256+k) = base[b,t,d] + (k+1)*rad[b,t,d]
//   sine = AMP * sin(2*pi*frac(phase))   (integer wrap shifts cancel in sin)
// noise_randn tile is staged into LDS via CDNA5 async-tensor copy:
//   global_load_async_to_lds_b128  +  s_wait_asynccnt  +  barrier
// ---------------------------------------------------------------------------
__global__ void __launch_bounds__(256)
sinegen_samples(const float* __restrict__ f0,
                const float* __restrict__ noise_randn,
                const double* __restrict__ base,
                const float* __restrict__ radf,
                float* __restrict__ out) {
    const int tile = blockIdx.x;        // 0 .. BATCH*TFRAMES-1
    const int b    = tile / TFRAMES;
    const int t    = tile % TFRAMES;
    const int k    = threadIdx.x;       // 0 .. 255 (sample within frame)
    const int s    = t * UPP + k;

    // ---- async-stage this block's noise tile (256 samples x 9) into LDS ----
    __shared__ float nbuf[UPP * DIM];   // 2304 floats = 9216 B = 576 x b128
    {
        const size_t gelem = ((size_t)b * SAMPLES + (size_t)t * UPP) * DIM;
        const uint64_t gbase = (uint64_t)(uintptr_t)(noise_randn + gelem); // 16B aligned
        const uint32_t lbase = (uint32_t)(uintptr_t)(&nbuf[0]);
        for (int j = k; j < (UPP * DIM) / 4; j += 256) {   // 576 b128 chunks
            uint64_t ga = gbase + (uint64_t)j * 16u;
            uint32_t la = lbase + (uint32_t)j * 16u;
            asm volatile("global_load_async_to_lds_b128 %0, %1, off"
                         :: "v"(la), "v"(ga) : "memory");
        }
        asm volatile("s_wait_asynccnt 0" ::: "memory");
    }
    __syncthreads();

    // ---- block-uniform frame data (compiler scalarizes these loads) ----
    const float f0v  = f0[b * TFRAMES + t];
    const float uvf  = (f0v > 0.0f) ? 1.0f : 0.0f;
    const float namp = (f0v > 0.0f) ? NOISE_STD : (float)(0.1 / 3.0);
    const int   fidx = (b * TFRAMES + t) * DIM;

    const size_t o_s = ((size_t)b * SAMPLES + (size_t)s) * DIM;
    out[UV_OFF + (size_t)b * SAMPLES + (size_t)s] = uvf;

    const double kk = (double)(k + 1);
#pragma unroll
    for (int d = 0; d < DIM; ++d) {
        double ph = base[fidx + d] + kk * (double)radf[fidx + d];
        double fr = ph - floor(ph);                       // frac -> [0,1)
        float  sv = AMP * sinf((float)(fr * 6.283185307179586));
        float  nz = namp * nbuf[k * DIM + d];             // conflict-free: stride 9 vs 64 banks
        out[o_s + d]             = sv * uvf + nz;         // sine_waves
        out[NOISE_OFF + o_s + d] = nz;                    // noise
    }
}

// ---------------------------------------------------------------------------
extern "C" void kernel_launch(void* const* d_in, const int* in_sizes, int n_in,
                              void* d_out, int out_size, void* d_ws, size_t ws_size,
                              hipStream_t stream) {
    const float* f0          = (const float*)d_in[0];   // [16,500]
    const float* rand_ini    = (const float*)d_in[1];   // [16,9]
    const float* noise_randn = (const float*)d_in[2];   // [16,128000,9]
    // d_in[3] = upp (256), fixed by config

    // workspace layout: f64 base[72000] | f32 rad[72000]  (864 KB, L2-resident)
    double* base = (double*)d_ws;
    float*  radf = (float*)((char*)d_ws + (size_t)BATCH * TFRAMES * DIM * sizeof(double));

    float* out = (float*)d_out;

    sinegen_frames<<<dim3(BATCH), dim3(32), 0, stream>>>(f0, rand_ini, base, radf);
    sinegen_samples<<<dim3(BATCH * TFRAMES), dim3(UPP), 0, stream>>>(
        f0, noise_randn, base, radf, out);
}